// EnhancedNoncommutativeKAOperator_53575422051073
// MI455X (gfx1250) — compile-verified
//
#include <hip/hip_runtime.h>

// ---------------------------------------------------------------------------
// EnhancedNoncommutativeKAOperator, MI455X (gfx1250).
// Output H is banded: |i-j| <= 11. Roofline: 67 MB store ~2.9us @ 23.3 TB/s;
// nonzero FLOPs ~ MFLOPs => pure store-bandwidth problem + tiny banded
// triple product A*H0*A^H done per 16-row tile with V_WMMA_F32_16X16X4_F32
// (A real => two real triple products for Re/Im planes). theta factored out
// of the f32 fragments (A~ = A/theta is O(1)); theta^2 applied in fp64.
// ---------------------------------------------------------------------------

#define DIMN   2048
#define PI_D   3.14159265358979323846
#define ZETA2D 1.6449340668482264365   // pi^2/6

typedef __attribute__((ext_vector_type(2))) float v2f;
typedef __attribute__((ext_vector_type(8))) float v8f;

// Workspace layout (in doubles): band tiles, frob partials, reg scalar.
// 128 row-tiles * 3 col-tiles * 16*16 * {re,im} = 196608 doubles (~1.57 MB).
#define WS_BAND 0
#define WS_PART (128 * 3 * 256 * 2)
#define WS_REG  (WS_PART + 128)

__device__ __forceinline__ bool is_prime_small(int n) {
  if (n < 2) return false;
  if (n < 4) return true;
  if ((n & 1) == 0) return false;
  for (int f = 3; f * f <= n; f += 2)
    if (n % f == 0) return false;
  return true;
}
// first 100 primes -> p <= 541 ; first 20 primes -> p <= 71
__device__ __forceinline__ bool corr_prime(int n) { return n <= 541 && is_prime_small(n); }
__device__ __forceinline__ bool alg_prime(int n)  { return n <= 71  && is_prime_small(n); }

// Arnold map divided by theta (theta-independent, O(1) entries, bandwidth 4).
__device__ __forceinline__ double atil(int i, int j) {
  if ((unsigned)i >= DIMN || (unsigned)j >= DIMN) return 0.0;
  int d = i - j; if (d < 0) d = -d;
  if (d > 4) return 0.0;
  double acc = 0.0;
  if (d == 0) {
#pragma unroll
    for (int sc = 1; sc <= 4; sc <<= 1)
      acc += cos(2.0 * PI_D * (double)i * (double)sc / (double)DIMN) / (double)sc;
  } else {
    int ii = (i < j) ? i : j;
#pragma unroll
    for (int sc = 1; sc <= 4; sc <<= 1)
      if (d <= sc)
        acc += exp(-(double)d / (10.0 * (double)sc)) *
               sin(PI_D * (double)(2 * ii + d) * (double)sc / (double)DIMN) /
               (double)sc;
  }
  return acc;
}

// Base KA operator entry H0[p][q] (diag n^-s + prime bands, bandwidth 3).
__device__ __forceinline__ void h0_entry(int p, int q, double sr, double si,
                                         double th, double* re, double* im) {
  *re = 0.0; *im = 0.0;
  if ((unsigned)p >= DIMN || (unsigned)q >= DIMN) return;
  int d = q - p;
  if (d == 0) {
    double ln  = log((double)(p + 1));
    double mag = exp(-sr * ln);
    *re =  mag * cos(si * ln);
    *im = -mag * sin(si * ln);
    if (corr_prime(p + 1)) *re += th * ln * (ZETA2D / (double)(p + 1));
  } else if (d > 0 && d <= 3) {
    if (corr_prime(p + 1)) *im =  th * log((double)(p + 1)) / (2.0 * (double)d);
  } else if (d < 0 && d >= -3) {
    if (corr_prime(q + 1)) *im = -th * log((double)(q + 1)) / (2.0 * (double)(-d));
  }
}

// Imaginary part of noncommutative-algebra entry (Hermitian, bandwidth 4).
__device__ __forceinline__ double alg_im(int i, int j, double th) {
  int d = j - i; int ad = d < 0 ? -d : d;
  if (ad == 0 || ad > 4) return 0.0;
  double tl = th;
  for (int l = 1; l < ad; ++l) tl *= th;
  double v = tl * exp(-(double)ad / 5.0);
  v = (d > 0) ? v : -v;
  if (ad == 1) {
    if (d > 0) { if (alg_prime(j)) v += th * log((double)j); }
    else       { if (alg_prime(i)) v -= th * log((double)i); }
  }
  return v;
}

// ---------------------------------------------------------------------------
// Kernel 1: one wave per 16-row tile. Computes the 16x48 band block of
// theta^2 * (A~ H0 A~^T) via chained V_WMMA_F32_16X16X4_F32, adds alg*|s|,
// stores band tiles (fp64) to workspace + per-block Frobenius partial.
// ---------------------------------------------------------------------------
__global__ __launch_bounds__(32) void band_triple_wmma(const double* __restrict__ pSr,
                                                       const double* __restrict__ pSi,
                                                       const double* __restrict__ pTh,
                                                       double* __restrict__ ws) {
  const int lane  = threadIdx.x;           // wave32
  const int blk   = blockIdx.x;            // 128 row tiles
  const int r0    = blk << 4;
  const int hrow0 = r0 - 8;                // H0 window rows  [r0-8,  r0+24)
  const int w0    = r0 - 24;               // H0 window cols  [r0-24, r0+40)
  const int lh    = lane >> 4;             // lane half (K-group select)
  const int l15   = lane & 15;

  const double sr  = pSr[0], si = pSi[0], th = pTh[0];
  const double th2 = th * th;
  const double smag = sqrt(sr * sr + si * si);

  __shared__ float  sH[2][32][64];   // H0 window, re/im planes (f32)
  __shared__ float  sT[16][64];      // T = A~ * H0 staging (per plane)
  __shared__ double sRed[32];

  // Fill H0 window analytically.
  for (int idx = lane; idx < 32 * 64; idx += 32) {
    int rw = idx >> 6, cl = idx & 63;
    double re, im;
    h0_entry(hrow0 + rw, w0 + cl, sr, si, th, &re, &im);
    sH[0][rw][cl] = (float)re;
    sH[1][rw][cl] = (float)im;
  }

  // Stage-1 A-fragments (16x4 f32 layout: lanes0-15 K={0,1}, lanes16-31 K={2,3}).
  v2f afr[8];
#pragma unroll
  for (int kk = 0; kk < 8; ++kk) {
    int k = 4 * kk + 2 * lh;
    afr[kk].x = (float)atil(r0 + l15, hrow0 + k);
    afr[kk].y = (float)atil(r0 + l15, hrow0 + k + 1);
  }

  v8f rr[2][3];
#pragma unroll
  for (int part = 0; part < 2; ++part) {
    // Stage 1: T(16x64) = A~(16x32) * H0(32x64), K accumulated in 4-chunks.
    v8f t[4];
#pragma unroll
    for (int ct = 0; ct < 4; ++ct)
      t[ct] = (v8f){0.f, 0.f, 0.f, 0.f, 0.f, 0.f, 0.f, 0.f};
#pragma unroll
    for (int kk = 0; kk < 8; ++kk) {
      v2f a = afr[kk];
      int kb = 4 * kk + 2 * lh;
#pragma unroll
      for (int ct = 0; ct < 4; ++ct) {
        v2f b;
        b.x = sH[part][kb + 0][16 * ct + l15];
        b.y = sH[part][kb + 1][16 * ct + l15];
        t[ct] = __builtin_amdgcn_wmma_f32_16x16x4_f32(false, a, false, b,
                                                      (short)0, t[ct], false, false);
      }
    }
    // Spill T (C-layout: M = v + 8*half, N = lane&15) to LDS row-major.
#pragma unroll
    for (int ct = 0; ct < 4; ++ct)
#pragma unroll
      for (int v = 0; v < 8; ++v)
        sT[v + 8 * lh][16 * ct + l15] = t[ct][v];

    // Stage 2: R(16x16) = T(16x64) * A~^T(64x16), three column tiles.
#pragma unroll
    for (int tt = 0; tt < 3; ++tt) {
      int c0 = r0 + 16 * (tt - 1);
      v8f r = (v8f){0.f, 0.f, 0.f, 0.f, 0.f, 0.f, 0.f, 0.f};
#pragma unroll
      for (int kk = 0; kk < 16; ++kk) {
        int k = 4 * kk + 2 * lh;
        v2f a, b;
        a.x = sT[l15][k];
        a.y = sT[l15][k + 1];
        b.x = (float)atil(c0 + l15, w0 + k);      // B[k][n] = A~[c0+n][w0+k]
        b.y = (float)atil(c0 + l15, w0 + k + 1);
        r = __builtin_amdgcn_wmma_f32_16x16x4_f32(false, a, false, b,
                                                  (short)0, r, false, false);
      }
      rr[part][tt] = r;
    }
  }

  // fp64 epilogue: scale by theta^2, add alg*|s|, write band tiles, frob sum.
  double sum = 0.0;
#pragma unroll
  for (int tt = 0; tt < 3; ++tt) {
    int c0 = r0 + 16 * (tt - 1);
#pragma unroll
    for (int v = 0; v < 8; ++v) {
      int M = v + 8 * lh, N = l15;
      int i = r0 + M, j = c0 + N;
      if ((unsigned)j < DIMN) {
        int d = i - j; int ad = d < 0 ? -d : d;
        double re = th2 * (double)rr[0][tt][v];
        double im = th2 * (double)rr[1][tt][v] + smag * alg_im(i, j, th);
        size_t o = ((size_t)(blk * 3 + tt) * 256 + (size_t)(M * 16 + N)) * 2;
        ws[WS_BAND + o]     = re;
        ws[WS_BAND + o + 1] = im;
        if (ad <= 11) sum += re * re + im * im;
      }
    }
  }
  sRed[lane] = sum;
  __builtin_amdgcn_wave_barrier();
  if (lane == 0) {                      // deterministic in-order reduction
    double s = 0.0;
    for (int k = 0; k < 32; ++k) s += sRed[k];
    ws[WS_PART + blk] = s;
  }
}

// ---------------------------------------------------------------------------
// Kernel 2: deterministic reduction of Frobenius partials -> reg scalar.
// ---------------------------------------------------------------------------
__global__ void frob_reduce(double* __restrict__ ws) {
  if (threadIdx.x == 0 && blockIdx.x == 0) {
    double s = 0.0;
    for (int b = 0; b < 128; ++b) s += ws[WS_PART + b];
    double rg = sqrt(s) * 1e-15;
    ws[WS_REG] = rg > 1e-18 ? rg : 1e-18;
  }
}

// ---------------------------------------------------------------------------
// Kernel 3: bandwidth-bound writer. One double2 (b128 store) per complex
// element: zeros outside |i-j|<=11, Hermitian-symmetrized band + reg*I inside.
// ---------------------------------------------------------------------------
__global__ __launch_bounds__(256) void finalize_store(const double* __restrict__ ws,
                                                      double* __restrict__ out) {
  size_t t = (size_t)blockIdx.x * 256 + threadIdx.x;   // one complex element
  int i = (int)(t >> 11);
  int j = (int)(t & 2047);
  double re = 0.0, im = 0.0;
  int d = i - j; int ad = d < 0 ? -d : d;
  if (ad <= 11) {
    int r1 = i >> 4, t1 = (j >> 4) - r1 + 1;           // always in [0,2]
    size_t o1 = ((size_t)(r1 * 3 + t1) * 256 + (size_t)((i & 15) * 16 + (j & 15))) * 2;
    int r2 = j >> 4, t2 = (i >> 4) - r2 + 1;
    size_t o2 = ((size_t)(r2 * 3 + t2) * 256 + (size_t)((j & 15) * 16 + (i & 15))) * 2;
    re = 0.5 * (ws[WS_BAND + o1]     + ws[WS_BAND + o2]);
    im = 0.5 * (ws[WS_BAND + o1 + 1] - ws[WS_BAND + o2 + 1]);   // 0.5*(H + H^dag)
    if (ad == 0) { re += ws[WS_REG]; im = 0.0; }
  }
  double2 val; val.x = re; val.y = im;
  reinterpret_cast<double2*>(out)[t] = val;            // 16B -> global_store_b128
}

extern "C" void kernel_launch(void* const* d_in, const int* in_sizes, int n_in,
                              void* d_out, int out_size, void* d_ws, size_t ws_size,
                              hipStream_t stream) {
  (void)in_sizes; (void)n_in; (void)out_size; (void)ws_size;
  const double* sR = (const double*)d_in[0];   // s_real (float64 scalar)
  const double* sI = (const double*)d_in[1];   // s_imag
  const double* th = (const double*)d_in[2];   // theta
  double* ws  = (double*)d_ws;                 // needs ~1.58 MB of scratch
  double* out = (double*)d_out;                // complex128 as (re,im) doubles

  band_triple_wmma<<<128, 32, 0, stream>>>(sR, sI, th, ws);
  frob_reduce<<<1, 32, 0, stream>>>(ws);
  finalize_store<<<(DIMN * DIMN) / 256, 256, 0, stream>>>(ws, out);
}